// IconCombo_41850161332740
// MI455X (gfx1250) — compile-verified
//
#include <hip/hip_runtime.h>

typedef __attribute__((ext_vector_type(2))) float v2f;
typedef __attribute__((ext_vector_type(8))) float v8f;

#define TN 4   // 16-wide N-tiles per wave (16x64 strip per wave)

// ---------------------------------------------------------------- build inv
__global__ void icw_build_inv(const int* __restrict__ n_id,
                              int* __restrict__ inv, int n) {
    int i = blockIdx.x * blockDim.x + threadIdx.x;
    if (i < n) inv[n_id[i]] = i;
}

// ---------------------------------------------------------------- out = bias
__global__ void icw_init_out(float* __restrict__ out,
                             const float* __restrict__ bias,
                             int n, int feat) {
    int idx = blockIdx.x * blockDim.x + threadIdx.x;     // float4 units
    int total = (n * feat) >> 2;
    if (idx < total) {
        int f4 = idx % (feat >> 2);
        ((float4*)out)[idx] = ((const float4*)bias)[f4];
    }
}

// ---------------------------------------------------------------- GEMM: hfeat = x @ W
// fp32 WMMA 16x16x4. One block per 16-row M tile; 8 waves, each wave computes
// a 16x(16*TN) strip. A-tile staged once into LDS per M tile. Per k-step:
// batch-issue all 2*TN B loads (one clause, MLP), then TN back-to-back WMMAs.
// B addressing strength-reduced: loop-invariant per-lane offsets + uniform
// scalar base bumped by 4*feat per k-step.
__global__ void __launch_bounds__(256)
icw_gemm_wmma(const float* __restrict__ x, const float* __restrict__ W,
              float* __restrict__ hfeat, int n, int feat) {
    extern __shared__ float As[];                // 16 * feat floats
    const int lane = threadIdx.x & 31;
    const int wave = __builtin_amdgcn_readfirstlane(threadIdx.x >> 5); // SGPR
    const int mt   = blockIdx.x;                 // M tile index

    // cooperative staging of A tile into LDS (coalesced float4)
    const int tot4 = (16 * feat) >> 2;
    for (int i = threadIdx.x; i < tot4; i += blockDim.x) {
        int elem = i << 2;
        int row  = elem / feat;
        int grow = mt * 16 + row;
        float4 v = make_float4(0.f, 0.f, 0.f, 0.f);
        if (grow < n)
            v = *(const float4*)(x + (size_t)grow * feat + (elem % feat));
        ((float4*)As)[i] = v;
    }
    __syncthreads();

    const int tn_total = feat >> 4;              // total 16-wide N tiles
    const int t0       = wave * TN;              // this wave's first N tile
    if (t0 >= tn_total) return;                  // scalar branch, after barrier

    // ISA 7.12.2: 32-bit A 16x4 -> lane<16: M=lane, K=k0+{0,1};
    //                              lane>=16: M=lane-16, K=k0+{2,3}
    const int mrow = lane & 15;
    const int kc   = (lane >> 4) * 2;

    // loop-invariant per-lane element offsets for the 2*TN B loads
    int off0[TN], off1[TN];
    #pragma unroll
    for (int t = 0; t < TN; ++t) {
        int tt = t0 + t;
        if (tt >= tn_total) tt = tn_total - 1;   // wave-uniform clamp
        int col = tt * 16 + (lane & 15);
        off0[t] = kc * feat + col;
        off1[t] = off0[t] + feat;
    }

    v8f c[TN];
    #pragma unroll
    for (int t = 0; t < TN; ++t) c[t] = (v8f){};

    const float* Wk = W;                         // uniform base, bumped per k
    const size_t kstride = (size_t)4 * feat;
    int as_base = mrow * feat + kc;              // LDS element index, bumped

    #pragma unroll 2
    for (int k0 = 0; k0 < feat; k0 += 4) {
        v2f a;
        a[0] = As[as_base];
        a[1] = As[as_base + 1];

        // issue ALL B loads for this k-step first (single clause, max MLP)
        v2f b[TN];
        #pragma unroll
        for (int t = 0; t < TN; ++t) {
            b[t][0] = Wk[off0[t]];
            b[t][1] = Wk[off1[t]];
        }

        // then TN back-to-back WMMAs (independent accumulators, no hazards)
        #pragma unroll
        for (int t = 0; t < TN; ++t) {
            // (neg_a, A, neg_b, B, c_mod, C, reuse_a, reuse_b)
            c[t] = __builtin_amdgcn_wmma_f32_16x16x4_f32(
                       false, a, false, b[t], (short)0, c[t], false, false);
        }

        Wk      += kstride;
        as_base += 4;
    }

    // C/D layout: lane<16 -> M = vgpr, lane>=16 -> M = vgpr + 8; N = lane&15
    const int nloc  = lane & 15;
    const int mbase = (lane >> 4) * 8;
    #pragma unroll
    for (int t = 0; t < TN; ++t) {
        if (t0 + t < tn_total) {                 // scalar guard (epilogue only)
            float* orow = hfeat + (size_t)mt * 16 * feat + (t0 + t) * 16;
            #pragma unroll
            for (int i = 0; i < 8; ++i) {
                int grow = mt * 16 + mbase + i;
                if (grow < n)
                    orow[(size_t)(mbase + i) * feat + nloc] = c[t][i];
            }
        }
    }
}

// ---------------------------------------------------------------- edge scatter
// One wave per edge. Lane owns feat/32 (=16) consecutive features.
// Edge header (indices, src/dst) is wave-uniform -> scalar path.
__global__ void __launch_bounds__(256)
icw_edge_scatter(const float* __restrict__ attns,   // (nets, E, H)
                 const int*   __restrict__ edges,   // (nets, 2, E)
                 const float* __restrict__ net_w,   // (H, nets)
                 const int*   __restrict__ inv,
                 const float* __restrict__ hfeat,   // (n, feat)
                 float*       __restrict__ out,     // (n, feat)
                 int nets, int E, int H, int D, int feat) {
    const int lane = threadIdx.x & 31;
    const int wave = __builtin_amdgcn_readfirstlane(threadIdx.x >> 5); // SGPR
    const long long e    = (long long)blockIdx.x * (blockDim.x >> 5) + wave;
    const long long etot = (long long)nets * E;
    if (e >= etot) return;                        // scalar branch

    const int net = (int)(e / E);
    const int idx = (int)(e % E);

    const int* eb  = edges + (size_t)net * 2 * E;
    const int  src = inv[eb[idx]];
    const int  dst = inv[eb[E + idx]];

    const float* arow = attns + ((size_t)net * E + idx) * H;
    const float* hrow = hfeat + (size_t)src * feat;
    float*       orow = out   + (size_t)dst * feat;

    const int fpl = feat >> 5;                   // floats per lane
    const int f0  = lane * fpl;
    #pragma unroll 4
    for (int f = f0; f < f0 + fpl; f += 4) {
        int   h     = f / D;
        float alpha = arow[h] * net_w[(size_t)h * nets + net];
        float4 v = *(const float4*)(hrow + f);
        atomicAdd(orow + f + 0, v.x * alpha);
        atomicAdd(orow + f + 1, v.y * alpha);
        atomicAdd(orow + f + 2, v.z * alpha);
        atomicAdd(orow + f + 3, v.w * alpha);
    }
}

// ---------------------------------------------------------------- launcher
extern "C" void kernel_launch(void* const* d_in, const int* in_sizes, int n_in,
                              void* d_out, int out_size, void* d_ws, size_t ws_size,
                              hipStream_t stream) {
    const float* x      = (const float*)d_in[0];
    const int*   n_id   = (const int*)  d_in[1];
    const float* attns  = (const float*)d_in[2];
    const int*   edges  = (const int*)  d_in[3];
    const float* net_w  = (const float*)d_in[4];
    const float* W      = (const float*)d_in[5];
    const float* bias   = (const float*)d_in[6];
    float*       out    = (float*)d_out;

    const int n    = in_sizes[1];                 // nodes
    const int feat = in_sizes[0] / n;             // H*D
    const long long a  = in_sizes[2];             // nets*E*H
    const long long b2 = in_sizes[3];             // nets*2*E
    const int H    = (int)((2 * a) / b2);
    const int nets = in_sizes[4] / H;             // (H, nets)
    const int E    = (int)(b2 / (2LL * nets));
    const int D    = feat / H;

    // workspace: [inv : n ints][hfeat : n*feat floats]
    int*   inv   = (int*)d_ws;
    float* hfeat = (float*)((char*)d_ws + (((size_t)n * 4 + 255) & ~(size_t)255));

    icw_build_inv<<<(n + 255) / 256, 256, 0, stream>>>(n_id, inv, n);

    const int out4 = (n * feat) >> 2;
    icw_init_out<<<(out4 + 255) / 256, 256, 0, stream>>>(out, bias, n, feat);

    icw_gemm_wmma<<<(n + 15) / 16, 256, (size_t)16 * feat * sizeof(float), stream>>>(
        x, W, hfeat, n, feat);

    const long long etot   = (long long)nets * E;
    const long long blocks = (etot + 7) / 8;      // 8 waves (edges) per block
    icw_edge_scatter<<<(unsigned)blocks, 256, 0, stream>>>(
        attns, edges, net_w, inv, hfeat, out, nets, E, H, D, feat);
}